// KerasCustomMappingLayer_7181185319050
// MI455X (gfx1250) — compile-verified
//
#include <hip/hip_runtime.h>
#include <hip/hip_bf16.h>
#include <stdint.h>

#define HH 512
#define BLK 64
#define XMAXF 512.0f
#define YMAXF 384.0f

// LDS byte-offset of a pointer into a __shared__ array (generic -> AS3 -> int).
#define LDS_OFF(p) \
    ((unsigned)(unsigned long long)(__attribute__((address_space(3))) const void*)(const void*)(p))

// One scan step for one k. All per-k params come from LDS tables.
__device__ __forceinline__ void map_step(
    float m0, float m1, float m2, float m3,
    const float4 t0k, const float4 t1k, const float2 flk,
    const bool nfse, float& px, float& py, float* __restrict__ o)
{
    const float lk = t0k.x, slnk = t0k.y, scos = t0k.z, ssin = t0k.w;
    const float wl = t1k.x, wr = t1k.y, wt = t1k.z, wb = t1k.w;
    const bool issl = (flk.x != 0.0f);
    const bool rr   = (flk.y != 0.0f);   // l > Y_MAX/2

    // normalize (vk, vk2) and (mv[H+k], mv[3H+k]) -> v_rsq_f32 + muls
    const float inv1 = __builtin_amdgcn_rsqf(m0 * m0 + m2 * m2);
    const float ck  = m0 * inv1;
    const float sk  = m2 * inv1;
    const float inv2 = __builtin_amdgcn_rsqf(m1 * m1 + m3 * m3);
    const float ck2 = m1 * inv2;
    const float sk2 = m3 * inv2;

    const float dx = lk * ck, dy = lk * sk;
    const float adx = __builtin_fabsf(dx), ady = __builtin_fabsf(dy);

    // x_delta = |dx|*[px<wl] - |dx|*[px>wr] + dx*[wl<px<wr]  (exclusive cases)
    const float xd = (px < wl) ? adx
                   : ((px > wr) ? -adx
                   : (((px > wl) && (px < wr)) ? dx : 0.0f));
    const float yd = (py < wt) ? ady
                   : ((py > wb) ? -ady
                   : (((py > wt) && (py < wb)) ? dy : 0.0f));

    const float _x = rr ? __builtin_fmaf(256.0f, m0, 256.0f) : (px + xd);
    const float _y = rr ? __builtin_fmaf(192.0f, m2, 192.0f) : (py + yd);

    const float oxn = _x * (1.0f / XMAXF);
    const float oyn = _y * (1.0f / YMAXF);

    o[0] = oxn;
    o[1] = oyn;
    if (issl) {                       // wave-uniform branch (same k across lanes)
        const float ex = __builtin_fmaf(ck2, slnk, _x);
        const float ey = __builtin_fmaf(sk2, slnk, _y);
        o[2] = ck2 * scos - sk2 * ssin;
        o[3] = ck2 * ssin + sk2 * scos;
        o[4] = ex * (1.0f / XMAXF);
        o[5] = ey * (1.0f / YMAXF);
        px = nfse ? ex : _x;
        py = nfse ? ey : _y;
    } else {
        o[2] = rr ? ck2 : ck;
        o[3] = rr ? sk2 : sk;
        o[4] = oxn;
        o[5] = oyn;
        px = _x;
        py = _y;
    }
}

__global__ __launch_bounds__(BLK)
void construct_map_kernel(const float* __restrict__ mv,     // (B, 4H)
                          const float* __restrict__ rel,    // (7, H)
                          const float* __restrict__ sp,     // (2,)
                          const float* __restrict__ lmp,    // (1,)
                          const int*   __restrict__ nfsep,  // (1,)
                          float* __restrict__ out,          // (B, H, 6)
                          int B)
{
    // Raw rel rows {0,1,3,4,5} staged by async DMA, then derived per-k tables.
    __shared__ float  relbuf[5 * HH];  // 10 KB
    __shared__ float4 t0[HH];          // {l_k, slider_len, slider_cos, slider_sin}
    __shared__ float4 t1[HH];          // {wall_l, wall_r, wall_t, wall_b}
    __shared__ float2 fl[HH];          // {is_slider, rerand}

    const int tid = (int)threadIdx.x;

    // ---- Stage rel rows into LDS with gfx1250 async copies (ASYNCcnt) ----
    // 5 rows x 128 float4 slots, BLK=64 threads -> exactly 2 slots/thread/row.
#pragma unroll
    for (int s = 0; s < 5; ++s) {
        const int srow = s + ((s >= 2) ? 1 : 0);   // rel rows 0,1,3,4,5
#pragma unroll
        for (int jj = 0; jj < (HH / 4) / BLK; ++jj) {
            const int j = jj * BLK + tid;
            const unsigned lds_off = LDS_OFF(&relbuf[s * HH + j * 4]);
            const unsigned long long ga =
                (unsigned long long)(uintptr_t)(rel + (size_t)srow * HH + (size_t)j * 4);
            asm volatile("global_load_async_to_lds_b128 %0, %1, off"
                         :
                         : "v"(lds_off), "v"(ga)
                         : "memory");
        }
    }
    asm volatile("s_wait_asynccnt 0x0" ::: "memory");
    __syncthreads();

    // ---- Build derived per-k tables from staged rel (8 entries/thread) ----
    const float lm = lmp[0];
#pragma unroll
    for (int kk2 = 0; kk2 < HH / BLK; ++kk2) {
        const int k = kk2 * BLK + tid;
        const float lk = lm * relbuf[4 * HH + k];           // l = lm * note_dist
        t0[k] = make_float4(lk, relbuf[1 * HH + k], relbuf[2 * HH + k], relbuf[3 * HH + k]);
        t1[k] = make_float4(__builtin_fmaf( 0.5f, lk, 0.05f * XMAXF),
                            __builtin_fmaf(-0.5f, lk, 0.95f * XMAXF),
                            __builtin_fmaf( 0.5f, lk, 0.05f * YMAXF),
                            __builtin_fmaf(-0.5f, lk, 0.95f * YMAXF));
        fl[k] = make_float2(relbuf[0 * HH + k], (lk > 0.5f * YMAXF) ? 1.0f : 0.0f);
    }
    __syncthreads();

    const int b = (int)(blockIdx.x * BLK) + tid;
    if (b >= B) return;

    const bool nfse = (nfsep[0] != 0);
    float px = sp[0];
    float py = sp[1];

    // Four sequential 128-bit input streams per lane + one output stream.
    const float4* __restrict__ p0 = (const float4*)(mv + (size_t)b * (4 * HH));
    const float4* __restrict__ p1 = p0 + (HH / 4);
    const float4* __restrict__ p2 = p0 + 2 * (HH / 4);
    const float4* __restrict__ p3 = p0 + 3 * (HH / 4);
    float4* __restrict__ po = (float4*)(out + (size_t)b * (HH * 6));

    // Warm-up: prefetch line 1 of each stream (line 0 is demand-missed next).
    __builtin_prefetch((const void*)(p0 + 8), 0, 1);
    __builtin_prefetch((const void*)(p1 + 8), 0, 1);
    __builtin_prefetch((const void*)(p2 + 8), 0, 1);
    __builtin_prefetch((const void*)(p3 + 8), 0, 1);

    // 16 chunks; each chunk consumes exactly one 128-B cacheline per stream.
    for (int c = 0; c < HH / 32; ++c) {
        if (c + 2 < HH / 32) {
            // gfx1250 global_prefetch_b8: two cachelines ahead on each stream.
            __builtin_prefetch((const void*)(p0 + (c + 2) * 8), 0, 1);
            __builtin_prefetch((const void*)(p1 + (c + 2) * 8), 0, 1);
            __builtin_prefetch((const void*)(p2 + (c + 2) * 8), 0, 1);
            __builtin_prefetch((const void*)(p3 + (c + 2) * 8), 0, 1);
        }
#pragma unroll
        for (int q = 0; q < 8; ++q) {
            const int kk = c * 8 + q;   // float4 index within streams
            const int k0 = kk * 4;      // scalar k base
            const float4 a0 = p0[kk];
            const float4 a1 = p1[kk];
            const float4 a2 = p2[kk];
            const float4 a3 = p3[kk];

            float r[24];
            map_step(a0.x, a1.x, a2.x, a3.x, t0[k0 + 0], t1[k0 + 0], fl[k0 + 0], nfse, px, py, r + 0);
            map_step(a0.y, a1.y, a2.y, a3.y, t0[k0 + 1], t1[k0 + 1], fl[k0 + 1], nfse, px, py, r + 6);
            map_step(a0.z, a1.z, a2.z, a3.z, t0[k0 + 2], t1[k0 + 2], fl[k0 + 2], nfse, px, py, r + 12);
            map_step(a0.w, a1.w, a2.w, a3.w, t0[k0 + 3], t1[k0 + 3], fl[k0 + 3], nfse, px, py, r + 18);

            float4* __restrict__ dst = po + kk * 6;
            dst[0] = make_float4(r[0],  r[1],  r[2],  r[3]);
            dst[1] = make_float4(r[4],  r[5],  r[6],  r[7]);
            dst[2] = make_float4(r[8],  r[9],  r[10], r[11]);
            dst[3] = make_float4(r[12], r[13], r[14], r[15]);
            dst[4] = make_float4(r[16], r[17], r[18], r[19]);
            dst[5] = make_float4(r[20], r[21], r[22], r[23]);
        }
    }
}

extern "C" void kernel_launch(void* const* d_in, const int* in_sizes, int n_in,
                              void* d_out, int out_size, void* d_ws, size_t ws_size,
                              hipStream_t stream) {
    const float* mv   = (const float*)d_in[0];  // mapvars (B, 4H) f32
    const float* rel  = (const float*)d_in[1];  // rel (7, H) f32
    const float* sp   = (const float*)d_in[2];  // start_pos (2,) f32
    const float* lmp  = (const float*)d_in[3];  // length_multiplier (1,) f32
    const int*   nfse = (const int*)d_in[4];    // next_from_slider_end scalar int
    float* out = (float*)d_out;                 // (B, H, 6) f32

    const int H = in_sizes[1] / 7;              // expected 512 (kernel assumes HH)
    const int B = in_sizes[0] / (4 * H);        // expected 32768
    (void)n_in; (void)out_size; (void)d_ws; (void)ws_size;

    // 64 threads = 2 wave32 waves per workgroup: spreads the fixed 1024-wave
    // budget (one lane per scan row) across 512 workgroups / many WGPs.
    const int grid = (B + BLK - 1) / BLK;
    construct_map_kernel<<<grid, BLK, 0, stream>>>(mv, rel, sp, lmp, nfse, out, B);
}